// LossRegressionGaussianNoCorrelations_54142357733402
// MI455X (gfx1250) — compile-verified
//
#include <hip/hip_runtime.h>
#include <math.h>

// CDNA5 wave32; WMMA f32 16x16x4: A = 16x4 f32 (2 VGPRs), B = 4x16 f32 (2 VGPRs), C/D = 16x16 f32 (8 VGPRs)
typedef __attribute__((ext_vector_type(2))) float v2f;
typedef __attribute__((ext_vector_type(4))) float v4f;   // native clang vector: OK for nontemporal builtin, loads as b128
typedef __attribute__((ext_vector_type(8))) float v8f;

#define LOG_SQRT_2PI 0.91893853320467274178f

// per-element: logq - logp = -log(sigma) - 0.5*((x-mu)/sigma)^2 + 0.5*x^2
// Fast paths: v_rcp-based divide (~2 ulp) and hardware v_log_f32 (~1 ulp on
// sigma in [0.5,1.5)). Per-element error ~1e-7 relative, uncorrelated, vs a
// ~1e7-magnitude fp32 sum -> far below fp32 reduction noise. Keeps the loop
// bandwidth-bound instead of VALU-bound.
__device__ __forceinline__ float elem_term(float x, float m, float s) {
    float z = __fdividef(x - m, s);
    return fmaf(0.5f * x, x, fmaf(-0.5f * z, z, -__logf(s)));
}

// Block-wide (256-thread) sum using V_WMMA_F32_16X16X4_F32 as the 64->16 column
// reducer. Every wave runs the WMMA with identical data and full EXEC (uniform
// control flow); each lane ends up holding the block total.
__device__ __forceinline__ float block_reduce_wmma(float acc, float* sdata) {
    const int tid = threadIdx.x;
    sdata[tid] = acc;
    __syncthreads();

    const int l = tid & 31;
    // Fold 256 partials -> 64 values (2 per lane) feeding the B matrix.
    // Any bijective placement into B is fine: we only need the grand total.
    float s0 = sdata[l]      + sdata[l + 64]  + sdata[l + 128] + sdata[l + 192];
    float s1 = sdata[l + 32] + sdata[l + 96]  + sdata[l + 160] + sdata[l + 224];

    v2f a; a[0] = 1.0f; a[1] = 1.0f;   // A = ones(16x4)
    v2f b; b[0] = s0;   b[1] = s1;     // B = 4x16 carrying the 64 partials
    v8f c = {};
    // D[i][j] = sum_k B[k][j]  (column sums, replicated over rows)
    v8f d = __builtin_amdgcn_wmma_f32_16x16x4_f32(
        /*neg_a=*/false, a, /*neg_b=*/false, b,
        /*c_mod=*/(short)0, c, /*reuse_a=*/false, /*reuse_b=*/false);

    // d[0]: lanes 0-15 hold the 16 column sums (row M=0); lanes 16-31 hold row
    // M=8 which is the identical column sums. XOR-reduce within 16-lane halves.
    float v = d[0];
    v += __shfl_xor(v, 8, 32);
    v += __shfl_xor(v, 4, 32);
    v += __shfl_xor(v, 2, 32);
    v += __shfl_xor(v, 1, 32);
    return v;   // every lane now holds the block total
}

__global__ void __launch_bounds__(256)
bnn_loss_partials(const float* __restrict__ x,
                  const float* __restrict__ mu,
                  const float* __restrict__ sg,
                  float* __restrict__ partials, int n) {
    __shared__ float sdata[256];
    const int tid = threadIdx.x;
    const unsigned gid = blockIdx.x * 256u + tid;
    const unsigned stride = gridDim.x * 256u;
    const unsigned n4 = (unsigned)(n >> 2);

    const v4f* __restrict__ x4 = (const v4f*)x;
    const v4f* __restrict__ m4 = (const v4f*)mu;
    const v4f* __restrict__ s4 = (const v4f*)sg;

    float acc = 0.0f;
    // Single-pass stream, zero reuse, working set (~201 MB) > L2 (192 MB):
    // non-temporal loads avoid thrashing L2.
    #pragma unroll 2
    for (unsigned i = gid; i < n4; i += stride) {
        v4f xv = __builtin_nontemporal_load(&x4[i]);
        v4f mv = __builtin_nontemporal_load(&m4[i]);
        v4f sv = __builtin_nontemporal_load(&s4[i]);
        acc += elem_term(xv[0], mv[0], sv[0]);
        acc += elem_term(xv[1], mv[1], sv[1]);
        acc += elem_term(xv[2], mv[2], sv[2]);
        acc += elem_term(xv[3], mv[3], sv[3]);
    }
    // Tail (n not multiple of 4) handled deterministically by global thread 0.
    if (gid == 0) {
        for (int i = (int)(n4 << 2); i < n; ++i) acc += elem_term(x[i], mu[i], sg[i]);
    }

    float tot = block_reduce_wmma(acc, sdata);
    if (tid == 0) partials[blockIdx.x] = tot;
}

__global__ void __launch_bounds__(256)
bnn_loss_final(const float* __restrict__ partials, int nparts,
               const float* __restrict__ mu_p,
               const float* __restrict__ sg_p,
               const float* __restrict__ y_true,
               float* __restrict__ out) {
    __shared__ float sdata[256];
    const int tid = threadIdx.x;
    float acc = 0.0f;
    for (int i = tid; i < nparts; i += 256) acc += partials[i];

    float tot = block_reduce_wmma(acc, sdata);
    if (tid == 0) {
        // Likelihood term is a scalar: keep it in precise IEEE math (one shot).
        float sp = sg_p[0];
        float z  = (y_true[0] - mu_p[0]) / sp;
        float loglik = -logf(sp) - LOG_SQRT_2PI - 0.5f * z * z;
        out[0] = tot - loglik;
    }
}

extern "C" void kernel_launch(void* const* d_in, const int* in_sizes, int n_in,
                              void* d_out, int out_size, void* d_ws, size_t ws_size,
                              hipStream_t stream) {
    const float* noisy = (const float*)d_in[0];
    const float* muw   = (const float*)d_in[1];
    const float* sgw   = (const float*)d_in[2];
    const float* mu_p  = (const float*)d_in[3];
    const float* sg_p  = (const float*)d_in[4];
    const float* ytru  = (const float*)d_in[5];
    float* out = (float*)d_out;
    float* partials = (float*)d_ws;

    const int n = in_sizes[0];
    const int n4 = n >> 2;
    int nblocks = (n4 + 255) / 256;
    if (nblocks > 1024) nblocks = 1024;   // ~64 fp32 elems/thread: deep tree, HBM-saturating MLP
    if (nblocks < 1) nblocks = 1;

    bnn_loss_partials<<<nblocks, 256, 0, stream>>>(noisy, muw, sgw, partials, n);
    bnn_loss_final<<<1, 256, 0, stream>>>(partials, nblocks, mu_p, sg_p, ytru, out);
}